// LinkPrediction_80530636800640
// MI455X (gfx1250) — compile-verified
//
#include <hip/hip_runtime.h>
#include <hip/hip_bf16.h>

typedef float v2f __attribute__((ext_vector_type(2)));
typedef float v8f __attribute__((ext_vector_type(8)));

#define D_FEAT 64
// K-paired W layout in LDS: Wp[(k/2)*WP_STRIDE + col*2 + (k&1)]
// Stride 160 dwords == 32 mod 64 banks -> lanes 0-15 (K pair 0,1) use banks
// 0..31, lanes 16-31 (K pair 2,3) use banks 32..63: conflict-free b64 reads.
#define WP_STRIDE 160
#define WP_SIZE (32 * WP_STRIDE)  // 32 K-pairs -> 5120 dwords = 20 KB

// ---------------------------------------------------------------------------
// Zero a float range.
// ---------------------------------------------------------------------------
__global__ void zero_kernel(float* __restrict__ p, long long n) {
  long long i = (long long)blockIdx.x * blockDim.x + threadIdx.x;
  if (i < n) p[i] = 0.0f;
}

// ---------------------------------------------------------------------------
// Edge scatter: neigh[dst] += feat[src]; deg[dst] += 1 (optional).
// 16 threads per edge, each thread owns 4 consecutive floats (float4 gather).
// Node table (25.6MB) is L2-resident on MI455X (192MB L2), so these atomics
// resolve in L2.
// ---------------------------------------------------------------------------
__global__ void scatter_kernel(const float* __restrict__ feat,
                               const int* __restrict__ src,
                               const int* __restrict__ dst,
                               float* __restrict__ neigh,
                               float* __restrict__ deg,
                               int n_edges, int add_deg) {
  long long gid = (long long)blockIdx.x * blockDim.x + threadIdx.x;
  int e = (int)(gid >> 4);
  int c = (int)(gid & 15);
  if (e >= n_edges) return;
  int s = src[e];
  int d = dst[e];
  const float4 v = *(const float4*)(feat + (long long)s * D_FEAT + c * 4);
  float* p = neigh + (long long)d * D_FEAT + c * 4;
  atomicAdd(p + 0, v.x);
  atomicAdd(p + 1, v.y);
  atomicAdd(p + 2, v.z);
  atomicAdd(p + 3, v.w);
  if (add_deg && c == 0) atomicAdd(deg + d, 1.0f);
}

// ---------------------------------------------------------------------------
// Fused SAGE layer: out = [relu]( ((neigh + self) / (deg+1)) @ W + b )
// One wave32 computes a 16-row x 64-col output tile using
// V_WMMA_F32_16X16X4_F32 (f32 A/B, f32 accum -> matches reference precision).
//
// 32-bit A layout (16x4, MxK):  lane&15 = M; lanes 0-15 hold K=0(v0),1(v1);
//                               lanes 16-31 hold K=2(v0),3(v1).
// 32-bit B layout (4x16, KxN):  mirrored with N = lane&15.
// 32-bit C/D layout (16x16):    N = lane&15; VGPR r holds M = r + 8*(lane>=16).
// W is staged in LDS in a K-paired layout so each B fragment is one aligned
// ds_load_b64 (no VGPR repacking) with conflict-free banking.
// ---------------------------------------------------------------------------
__global__ void sage_gemm_kernel(const float* __restrict__ neigh,
                                 const float* __restrict__ self,
                                 const float* __restrict__ deg,
                                 const float* __restrict__ W,     // [64][64]
                                 const float* __restrict__ bias,  // [64]
                                 float* __restrict__ out,
                                 int n_nodes, int do_relu) {
  __shared__ float Wp[WP_SIZE];  // 20 KB, K-paired + bank-padded
  for (int i = threadIdx.x; i < D_FEAT * D_FEAT; i += blockDim.x) {
    const int k = i >> 6;
    const int col = i & 63;
    Wp[(k >> 1) * WP_STRIDE + col * 2 + (k & 1)] = W[i];
  }
  __syncthreads();

  const int lane = threadIdx.x & 31;
  const int waveInBlock = threadIdx.x >> 5;
  const int waveId = blockIdx.x * (blockDim.x >> 5) + waveInBlock;
  const int rowBase = waveId * 16;
  if (rowBase >= n_nodes) return;  // uniform across wave (EXEC stays all-1s)

  const int halfSel = lane >> 4;  // 0: K pair {0,1}, 1: K pair {2,3}
  const int l15 = lane & 15;
  const int row = rowBase + l15;  // A-matrix row for this lane
  const float inv = 1.0f / (deg[row] + 1.0f);

  v8f acc[4] = {};  // 4 N-tiles of 16 -> 64 output columns

  const float* nrow = neigh + (long long)row * D_FEAT;
  const float* srow = self + (long long)row * D_FEAT;

#pragma unroll
  for (int kt = 0; kt < 16; ++kt) {  // K = 64, 4 per WMMA
    const int k = kt * 4 + halfSel * 2;  // even
    const float2 ng = *(const float2*)(nrow + k);
    const float2 sf = *(const float2*)(srow + k);
    v2f a;
    a.x = (ng.x + sf.x) * inv;
    a.y = (ng.y + sf.y) * inv;
    const float* bbase = &Wp[(k >> 1) * WP_STRIDE];
#pragma unroll
    for (int nt = 0; nt < 4; ++nt) {
      const int col = nt * 16 + l15;
      const v2f b = *(const v2f*)(bbase + col * 2);  // {W[k][col], W[k+1][col]}
      acc[nt] = __builtin_amdgcn_wmma_f32_16x16x4_f32(
          /*neg_a=*/false, a, /*neg_b=*/false, b,
          /*c_mod=*/(short)0, acc[nt], /*reuse_a=*/false, /*reuse_b=*/false);
    }
  }

  // Epilogue: bias (+ relu), scatter to the 16x64 tile
  const int rOff = halfSel * 8;
#pragma unroll
  for (int nt = 0; nt < 4; ++nt) {
    const int col = nt * 16 + l15;
    const float bv = bias[col];
#pragma unroll
    for (int r = 0; r < 8; ++r) {
      float v = acc[nt][r] + bv;
      if (do_relu) v = fmaxf(v, 0.0f);
      out[(long long)(rowBase + rOff + r) * D_FEAT + col] = v;
    }
  }
}

// ---------------------------------------------------------------------------
// Pair scoring: accum += sum_i -log_sigmoid(sign * dot(emb[a_i], emb[b_i]))
// Stable: -logsig(x) = log1p(exp(-|x|)) - min(x, 0)
// wave32 shuffle reduction, then cross-wave LDS pass, 1 atomic per block.
// ---------------------------------------------------------------------------
__global__ void score_kernel(const float* __restrict__ emb,
                             const int* __restrict__ a_idx,
                             const int* __restrict__ b_idx,
                             int n_pairs, float sign,
                             float* __restrict__ accum) {
  __shared__ float red[8];  // 256 threads = 8 waves (wave32)
  int i = blockIdx.x * blockDim.x + threadIdx.x;
  float val = 0.0f;
  if (i < n_pairs) {
    const float4* pa = (const float4*)(emb + (long long)a_idx[i] * D_FEAT);
    const float4* pb = (const float4*)(emb + (long long)b_idx[i] * D_FEAT);
    float dot = 0.0f;
#pragma unroll
    for (int j = 0; j < D_FEAT / 4; ++j) {
      float4 x = pa[j];
      float4 y = pb[j];
      dot += x.x * y.x + x.y * y.y + x.z * y.z + x.w * y.w;
    }
    float s = sign * dot;
    val = log1pf(expf(-fabsf(s))) - fminf(s, 0.0f);
  }
#pragma unroll
  for (int off = 16; off > 0; off >>= 1) val += __shfl_down(val, off, 32);
  const int lane = threadIdx.x & 31;
  const int wave = threadIdx.x >> 5;
  if (lane == 0) red[wave] = val;
  __syncthreads();
  if (wave == 0) {
    float v = (lane < 8) ? red[lane] : 0.0f;
#pragma unroll
    for (int off = 4; off > 0; off >>= 1) v += __shfl_down(v, off, 32);
    if (lane == 0) atomicAdd(accum, v);
  }
}

__global__ void finalize_kernel(const float* __restrict__ accum,
                                float inv_epos, float* __restrict__ out) {
  if (threadIdx.x == 0 && blockIdx.x == 0) out[0] = accum[0] * inv_epos;
}

// ---------------------------------------------------------------------------
// Launch
// ---------------------------------------------------------------------------
extern "C" void kernel_launch(void* const* d_in, const int* in_sizes, int n_in,
                              void* d_out, int out_size, void* d_ws,
                              size_t ws_size, hipStream_t stream) {
  const float* features = (const float*)d_in[0];
  const float* W1 = (const float*)d_in[1];
  const float* b1 = (const float*)d_in[2];
  const float* W2 = (const float*)d_in[3];
  const float* b2 = (const float*)d_in[4];
  const int* src = (const int*)d_in[5];
  const int* dst = (const int*)d_in[6];
  const int* pos_src = (const int*)d_in[7];
  const int* pos_dst = (const int*)d_in[8];
  const int* neg_src = (const int*)d_in[9];
  const int* neg_dst = (const int*)d_in[10];

  const int n_nodes = in_sizes[0] / D_FEAT;
  const int n_edges = in_sizes[5];
  const int e_pos = in_sizes[7];
  const int n_neg = in_sizes[9];
  const long long ND = (long long)n_nodes * D_FEAT;

  // Workspace layout (floats): [neigh ND][deg N][accum 1][h ND][emb ND]
  float* neigh = (float*)d_ws;
  float* deg = neigh + ND;
  float* accum = deg + n_nodes;
  float* h = accum + 1;
  float* emb = h + ND;

  const int B = 256;
  const long long tscatter = (long long)n_edges * 16;
  const int scatter_blocks = (int)((tscatter + B - 1) / B);
  const int waves = (n_nodes + 15) / 16;
  const int gemm_blocks = (waves + 7) / 8;  // 8 waves per 256-thread block

  // Layer 1: zero (neigh+deg+accum contiguous), scatter, gemm+relu
  {
    long long n = ND + n_nodes + 1;
    zero_kernel<<<(int)((n + B - 1) / B), B, 0, stream>>>(neigh, n);
  }
  scatter_kernel<<<scatter_blocks, B, 0, stream>>>(features, src, dst, neigh,
                                                   deg, n_edges, /*deg=*/1);
  sage_gemm_kernel<<<gemm_blocks, B, 0, stream>>>(neigh, features, deg, W1, b1,
                                                  h, n_nodes, /*relu=*/1);

  // Layer 2: re-zero neigh only (deg is identical both layers)
  zero_kernel<<<(int)((ND + B - 1) / B), B, 0, stream>>>(neigh, ND);
  scatter_kernel<<<scatter_blocks, B, 0, stream>>>(h, src, dst, neigh, deg,
                                                   n_edges, /*deg=*/0);
  sage_gemm_kernel<<<gemm_blocks, B, 0, stream>>>(neigh, h, deg, W2, b2, emb,
                                                  n_nodes, /*relu=*/0);

  // Scoring: pos uses +score, neg uses -score
  score_kernel<<<(e_pos + B - 1) / B, B, 0, stream>>>(emb, pos_src, pos_dst,
                                                      e_pos, 1.0f, accum);
  score_kernel<<<(n_neg + B - 1) / B, B, 0, stream>>>(emb, neg_src, neg_dst,
                                                      n_neg, -1.0f, accum);

  finalize_kernel<<<1, 32, 0, stream>>>(accum, 1.0f / (float)e_pos,
                                        (float*)d_out);
}